// OnlineTripletLoss_6511170421616
// MI455X (gfx1250) — compile-verified
//
#include <hip/hip_runtime.h>
#include <hip/hip_bf16.h>

typedef __attribute__((ext_vector_type(2))) float v2f;
typedef __attribute__((ext_vector_type(4))) float v4f;
typedef __attribute__((ext_vector_type(8))) float v8f;
typedef __attribute__((ext_vector_type(4))) int   v4i;

#define NB        8192          // batch rows
#define DDIM      512           // feature dim (K)
#define KC        64            // k-chunk staged in LDS (double buffered)
#define CSTRIDE   (KC + 4)      // LDS row stride (bank step 4 -> conflict-free b64)
#define NCHUNK    (DDIM / KC)   // 8 chunks per panel
#define PANEL     64            // columns per panel (4 WMMA tiles per wave)
#define ROWS_WG   128           // rows per workgroup (8 waves x 16 rows)
#define CSPLIT    16            // column splits across grid.y
#define PANELS_PER_SPLIT ((NB / PANEL) / CSPLIT)   // 8 panels -> 512 cols per split

// ---------------------------------------------------------------------------
// Async copy global -> LDS (16B per lane), tracked by ASYNCcnt.
// Builtin expects: (int4 addrspace(1)*, int4 addrspace(3)*, imm offset, cpol)
// ---------------------------------------------------------------------------
typedef __attribute__((address_space(1))) v4i as1_v4i;
typedef __attribute__((address_space(3))) v4i as3_v4i;

__device__ __forceinline__ void async_cp16(const float* g, float* l) {
#if __has_builtin(__builtin_amdgcn_global_load_async_to_lds_b128)
    __builtin_amdgcn_global_load_async_to_lds_b128(
        (as1_v4i*)(uintptr_t)g,
        (as3_v4i*)(unsigned)(uintptr_t)l,   // generic LDS addr truncates to AS3 offset
        0, 0);
#else
    *(v4f*)l = *(const v4f*)g;              // synchronous fallback
#endif
}

__device__ __forceinline__ void wait_async0() {
#if __has_builtin(__builtin_amdgcn_s_wait_asynccnt)
    __builtin_amdgcn_s_wait_asynccnt(0);
#else
    asm volatile("s_wait_asynccnt 0x0" ::: "memory");
#endif
}

// ---------------------------------------------------------------------------
// Kernel 1: row-normalize anchor/positive, compute cos_ap. One wave per row.
// ---------------------------------------------------------------------------
__global__ __launch_bounds__(256) void normalize_kernel(
    const float* __restrict__ anchor, const float* __restrict__ positive,
    float* __restrict__ a_n, float* __restrict__ p_n, float* __restrict__ cos_ap)
{
    const int wave = threadIdx.x >> 5;
    const int lane = threadIdx.x & 31;
    const int row  = blockIdx.x * 8 + wave;

    const float* ar = anchor   + (size_t)row * DDIM;
    const float* pr = positive + (size_t)row * DDIM;

    v4f av[4], pv[4];
    float sa = 0.f, sp = 0.f, sap = 0.f;
#pragma unroll
    for (int j = 0; j < 4; ++j) {
        const int off = lane * 4 + j * 128;          // float4 index lane + j*32
        av[j] = *(const v4f*)(ar + off);
        pv[j] = *(const v4f*)(pr + off);
        sa  += av[j].x*av[j].x + av[j].y*av[j].y + av[j].z*av[j].z + av[j].w*av[j].w;
        sp  += pv[j].x*pv[j].x + pv[j].y*pv[j].y + pv[j].z*pv[j].z + pv[j].w*pv[j].w;
        sap += av[j].x*pv[j].x + av[j].y*pv[j].y + av[j].z*pv[j].z + av[j].w*pv[j].w;
    }
#pragma unroll
    for (int m = 16; m >= 1; m >>= 1) {
        sa  += __shfl_xor(sa,  m, 32);
        sp  += __shfl_xor(sp,  m, 32);
        sap += __shfl_xor(sap, m, 32);
    }
    const float na = sqrtf(sa), np = sqrtf(sp);
    const float ra = 1.0f / na, rp = 1.0f / np;

    float* arow = a_n + (size_t)row * DDIM;
    float* prow = p_n + (size_t)row * DDIM;
#pragma unroll
    for (int j = 0; j < 4; ++j) {
        const int off = lane * 4 + j * 128;
        v4f an = { av[j].x*ra, av[j].y*ra, av[j].z*ra, av[j].w*ra };
        v4f pn = { pv[j].x*rp, pv[j].y*rp, pv[j].z*rp, pv[j].w*rp };
        *(v4f*)(arow + off) = an;
        *(v4f*)(prow + off) = pn;
    }
    if (lane == 0)
        cos_ap[row] = sap / fmaxf(na * np, 1e-8f);
}

// ---------------------------------------------------------------------------
// Kernel 2: fused sim = a_n @ p_n^T (fp32 WMMA 16x16x4) + per-row running
// argmax of |sim-1| (diag & score==0 masked), first-index tie-break.
// p_n panel k-chunks staged via async global->LDS copies, double buffered.
// ---------------------------------------------------------------------------
extern __shared__ float p_buf[];   // [2][PANEL][CSTRIDE] fp32 = 34816 bytes

__global__ __launch_bounds__(256) void gemm_argmax_kernel(
    const float* __restrict__ a_n, const float* __restrict__ p_n,
    float* __restrict__ candScore, float* __restrict__ candSim,
    int* __restrict__ candCol)
{
    const int tid  = threadIdx.x;
    const int wave = tid >> 5;
    const int lane = tid & 31;
    const int l15  = lane & 15;
    const int hi   = lane >> 4;          // half-wave select (k offset 2*hi)

    const int row_base = blockIdx.x * ROWS_WG + wave * 16;
    const int split    = blockIdx.y;

    // A fragment base: lane l -> row (row_base + l15), k offset 2*hi
    const float* a_row = a_n + (size_t)(row_base + l15) * DDIM + 2 * hi;

    float bestScore[8], bestSim[8];
    int   bestCol[8];
#pragma unroll
    for (int v = 0; v < 8; ++v) {
        bestScore[v] = -__builtin_inff();
        bestSim[v]   = 0.f;
        bestCol[v]   = 0x7fffffff;
    }

    int cur = 0;
    for (int p = 0; p < PANELS_PER_SPLIT; ++p) {
        const int panel    = split * PANELS_PER_SPLIT + p;
        const int col_base = panel * PANEL;

        // stage chunk 0 of this panel into buffer `cur`
        {
            float* dst = p_buf + cur * (PANEL * CSTRIDE);
            for (int i = tid; i < PANEL * (KC / 4); i += 256) {
                const int c  = i >> 4;              // KC/4 = 16 float4 per row
                const int kk = (i & 15) * 4;
                async_cp16(p_n + (size_t)(col_base + c) * DDIM + 0 + kk,
                           dst + c * CSTRIDE + kk);
            }
        }

        v8f acc0 = {}, acc1 = {}, acc2 = {}, acc3 = {};

        for (int ch = 0; ch < NCHUNK; ++ch) {
            wait_async0();
            __syncthreads();             // buffer `cur` ready for all waves

            // prefetch next chunk into the other buffer (overlaps compute)
            if (ch + 1 < NCHUNK) {
                float* dst = p_buf + (cur ^ 1) * (PANEL * CSTRIDE);
                const int kc_next = (ch + 1) * KC;
                for (int i = tid; i < PANEL * (KC / 4); i += 256) {
                    const int c  = i >> 4;
                    const int kk = (i & 15) * 4;
                    async_cp16(p_n + (size_t)(col_base + c) * DDIM + kc_next + kk,
                               dst + c * CSTRIDE + kk);
                }
            }

            const float* buf = p_buf + cur * (PANEL * CSTRIDE);
            const int kc = ch * KC;
#pragma unroll 4
            for (int k = 0; k < KC; k += 4) {
                const v2f af = *(const v2f*)(a_row + kc + k);
                const float* pb = buf + l15 * CSTRIDE + k + 2 * hi;
                const v2f b0 = *(const v2f*)(pb);
                const v2f b1 = *(const v2f*)(pb + 16 * CSTRIDE);
                const v2f b2 = *(const v2f*)(pb + 32 * CSTRIDE);
                const v2f b3 = *(const v2f*)(pb + 48 * CSTRIDE);
                acc0 = __builtin_amdgcn_wmma_f32_16x16x4_f32(false, af, false, b0, (short)0, acc0, false, false);
                acc1 = __builtin_amdgcn_wmma_f32_16x16x4_f32(false, af, false, b1, (short)0, acc1, false, false);
                acc2 = __builtin_amdgcn_wmma_f32_16x16x4_f32(false, af, false, b2, (short)0, acc2, false, false);
                acc3 = __builtin_amdgcn_wmma_f32_16x16x4_f32(false, af, false, b3, (short)0, acc3, false, false);
            }
            cur ^= 1;
        }

        // In-register per-row argmax across this panel's 64 columns.
        // Element: acc_t[v] = sim[row_base + v + 8*hi][col_base + t*16 + l15]
#pragma unroll
        for (int t = 0; t < 4; ++t) {
            const v8f a = (t == 0) ? acc0 : (t == 1) ? acc1 : (t == 2) ? acc2 : acc3;
            const int gcol = col_base + t * 16 + l15;
#pragma unroll
            for (int v = 0; v < 8; ++v) {
                const int grow = row_base + v + 8 * hi;
                float s  = a[v];
                float sc = fabsf(s - 1.0f);
                if (gcol == grow || sc == 0.0f) sc = -__builtin_inff();
                float bs = s;
                int   bc = gcol;
                // butterfly over the 16 lanes of this half-wave
#pragma unroll
                for (int m = 1; m <= 8; m <<= 1) {
                    const float sc2 = __shfl_xor(sc, m, 32);
                    const float bs2 = __shfl_xor(bs, m, 32);
                    const int   bc2 = __shfl_xor(bc, m, 32);
                    if (sc2 > sc || (sc2 == sc && bc2 < bc)) { sc = sc2; bs = bs2; bc = bc2; }
                }
                if (sc > bestScore[v] || (sc == bestScore[v] && bc < bestCol[v])) {
                    bestScore[v] = sc; bestSim[v] = bs; bestCol[v] = bc;
                }
            }
        }
    }

    // lanes 0 and 16 hold the reduced result for rows v (+8 for hi half)
    if (l15 == 0) {
#pragma unroll
        for (int v = 0; v < 8; ++v) {
            const int row = row_base + v + 8 * hi;
            const int ci  = row * CSPLIT + split;
            candScore[ci] = bestScore[v];
            candSim[ci]   = bestSim[v];
            candCol[ci]   = bestCol[v];
        }
    }
}

// ---------------------------------------------------------------------------
// Kernel 3: combine per-split candidates, compute mean triplet loss.
// ---------------------------------------------------------------------------
__global__ __launch_bounds__(256) void loss_kernel(
    const float* __restrict__ cos_ap,
    const float* __restrict__ candScore, const float* __restrict__ candSim,
    const int* __restrict__ candCol, float* __restrict__ out)
{
    __shared__ float red[256];
    float acc = 0.f;
    for (int row = threadIdx.x; row < NB; row += 256) {
        float bsc = -__builtin_inff(), bsim = 0.f;
        int bcol = 0x7fffffff;
        for (int s = 0; s < CSPLIT; ++s) {          // splits in ascending col order
            const int ci = row * CSPLIT + s;
            const float sc = candScore[ci];
            const int   cc = candCol[ci];
            if (sc > bsc || (sc == bsc && cc < bcol)) {
                bsc = sc; bsim = candSim[ci]; bcol = cc;
            }
        }
        const float loss = 1.0f + cos_ap[row] - bsim;   // MARGIN=1, cos_an = bsim
        acc += loss > 0.f ? loss : 0.f;
    }
    red[threadIdx.x] = acc;
    __syncthreads();
    for (int off = 128; off > 0; off >>= 1) {
        if (threadIdx.x < off) red[threadIdx.x] += red[threadIdx.x + off];
        __syncthreads();
    }
    if (threadIdx.x == 0) out[0] = red[0] / (float)NB;
}

// ---------------------------------------------------------------------------
extern "C" void kernel_launch(void* const* d_in, const int* in_sizes, int n_in,
                              void* d_out, int out_size, void* d_ws, size_t ws_size,
                              hipStream_t stream) {
    const float* anchor   = (const float*)d_in[0];
    const float* positive = (const float*)d_in[1];
    float* out = (float*)d_out;

    // workspace layout (floats): a_n | p_n | cos_ap | candScore | candSim | candCol
    float* ws      = (float*)d_ws;
    float* a_n     = ws;
    float* p_n     = a_n + (size_t)NB * DDIM;
    float* cos_ap  = p_n + (size_t)NB * DDIM;
    float* cScore  = cos_ap + NB;
    float* cSim    = cScore + (size_t)NB * CSPLIT;
    int*   cCol    = (int*)(cSim + (size_t)NB * CSPLIT);

    normalize_kernel<<<NB / 8, 256, 0, stream>>>(anchor, positive, a_n, p_n, cos_ap);

    dim3 grid(NB / ROWS_WG, CSPLIT);
    size_t lds = (size_t)2 * PANEL * CSTRIDE * sizeof(float);   // 34816 B
    gemm_argmax_kernel<<<grid, 256, lds, stream>>>(a_n, p_n, cScore, cSim, cCol);

    loss_kernel<<<1, 256, 0, stream>>>(cos_ap, cScore, cSim, cCol, out);
}